// WIPERAttention_36584531428189
// MI455X (gfx1250) — compile-verified
//
#include <hip/hip_runtime.h>
#include <math.h>

typedef float v2f __attribute__((ext_vector_type(2)));
typedef float v8f __attribute__((ext_vector_type(8)));

#define D_MODEL 1024
#define SEQ     1024
#define BATCH   2
#define HEADS   16
#define DH      64
#define M_TOT   (BATCH * SEQ)               // 2048
#define BHS     (BATCH * HEADS * SEQ)       // 32768
#define NQ      (BATCH * HEADS * SEQ * DH)  // 2,097,152 floats per tensor

#define WMMA_F32(A, B, C) \
    __builtin_amdgcn_wmma_f32_16x16x4_f32(false, (A), false, (B), (short)0, (C), false, false)

__constant__ float c_tau[12] = {0.05f, 0.05f, 0.05f, 0.15f, 0.25f, 0.35f,
                                0.45f, 0.55f, 0.65f, 0.70f, 0.75f, 0.80f};

// ---------------------------------------------------------------------------
// Pipelined fp32 WMMA GEMM body: one wave computes a 32(M) x 64(N) tile of
// C = A(M x 1024) * W^T (W is N x 1024, row-major).  8 accumulators; the
// next K-group's 6 fragments are loaded before the current 8 WMMAs issue, so
// loads overlap the matrix pipe instead of s_wait_loadcnt 0 per WMMA.
// ---------------------------------------------------------------------------
__device__ __forceinline__ void gemm32x64_pipe(
    const float* __restrict__ a0p, const float* __restrict__ a1p,
    const float* __restrict__ b0p, const float* __restrict__ b1p,
    const float* __restrict__ b2p, const float* __restrict__ b3p,
    v8f acc[8])
{
    v2f a0 = *(const v2f*)(a0p);
    v2f a1 = *(const v2f*)(a1p);
    v2f f0 = *(const v2f*)(b0p);
    v2f f1 = *(const v2f*)(b1p);
    v2f f2 = *(const v2f*)(b2p);
    v2f f3 = *(const v2f*)(b3p);

    #pragma unroll 4
    for (int k0 = 0; k0 < D_MODEL - 4; k0 += 4) {
        const int kn = k0 + 4;
        v2f na0 = *(const v2f*)(a0p + kn);
        v2f na1 = *(const v2f*)(a1p + kn);
        v2f nf0 = *(const v2f*)(b0p + kn);
        v2f nf1 = *(const v2f*)(b1p + kn);
        v2f nf2 = *(const v2f*)(b2p + kn);
        v2f nf3 = *(const v2f*)(b3p + kn);
        acc[0] = WMMA_F32(a0, f0, acc[0]);
        acc[1] = WMMA_F32(a0, f1, acc[1]);
        acc[2] = WMMA_F32(a0, f2, acc[2]);
        acc[3] = WMMA_F32(a0, f3, acc[3]);
        acc[4] = WMMA_F32(a1, f0, acc[4]);
        acc[5] = WMMA_F32(a1, f1, acc[5]);
        acc[6] = WMMA_F32(a1, f2, acc[6]);
        acc[7] = WMMA_F32(a1, f3, acc[7]);
        a0 = na0; a1 = na1; f0 = nf0; f1 = nf1; f2 = nf2; f3 = nf3;
    }
    acc[0] = WMMA_F32(a0, f0, acc[0]);
    acc[1] = WMMA_F32(a0, f1, acc[1]);
    acc[2] = WMMA_F32(a0, f2, acc[2]);
    acc[3] = WMMA_F32(a0, f3, acc[3]);
    acc[4] = WMMA_F32(a1, f0, acc[4]);
    acc[5] = WMMA_F32(a1, f1, acc[5]);
    acc[6] = WMMA_F32(a1, f2, acc[6]);
    acc[7] = WMMA_F32(a1, f3, acc[7]);
}

// ---------------------------------------------------------------------------
// Kernel 1: fused QKV projection, scattered into (B,H,S,Dh).
// ---------------------------------------------------------------------------
__global__ void __launch_bounds__(32)
k_qkv_gemm(const float* __restrict__ x,
           const float* __restrict__ Wq,
           const float* __restrict__ Wk,
           const float* __restrict__ Wv,
           float* __restrict__ qkv_ws)
{
    const int lane = threadIdx.x & 31;
    const int half = lane >> 4;
    const int lm   = lane & 15;
    const int m0   = blockIdx.x * 32;
    const int n0   = blockIdx.y * 64;
    const int which = blockIdx.z;

    const float* __restrict__ W = (which == 0) ? Wq : ((which == 1) ? Wk : Wv);
    float* __restrict__ outp = qkv_ws + (size_t)which * NQ;

    const float* a0p = x + (size_t)(m0 +  0 + lm) * D_MODEL + 2 * half;
    const float* a1p = x + (size_t)(m0 + 16 + lm) * D_MODEL + 2 * half;
    const float* b0p = W + (size_t)(n0 +  0 + lm) * D_MODEL + 2 * half;
    const float* b1p = W + (size_t)(n0 + 16 + lm) * D_MODEL + 2 * half;
    const float* b2p = W + (size_t)(n0 + 32 + lm) * D_MODEL + 2 * half;
    const float* b3p = W + (size_t)(n0 + 48 + lm) * D_MODEL + 2 * half;

    v8f acc[8] = {};
    gemm32x64_pipe(a0p, a1p, b0p, b1p, b2p, b3p, acc);

    const int b = m0 >> 10;
    const int s = m0 & 1023;
    #pragma unroll
    for (int r = 0; r < 2; ++r) {
        #pragma unroll
        for (int j = 0; j < 8; ++j) {
            const int srow = s + r * 16 + j + 8 * half;
            #pragma unroll
            for (int t = 0; t < 4; ++t) {
                const int n = n0 + t * 16 + lm;
                const int h = n >> 6, d = n & 63;
                outp[(((size_t)(b * HEADS + h) * SEQ + srow) * DH) + d] = acc[r * 4 + t][j];
            }
        }
    }
}

// ---------------------------------------------------------------------------
// Kernel 2: per-key saliency  sal = min(||K||^2 / Dh, 2)
// ---------------------------------------------------------------------------
__global__ void k_sal(const float* __restrict__ kws, float* __restrict__ sal)
{
    const int i = blockIdx.x * blockDim.x + threadIdx.x;
    if (i >= BHS) return;
    const float4* kp = (const float4*)(kws + (size_t)i * DH);
    float acc = 0.f;
    #pragma unroll
    for (int d = 0; d < DH / 4; ++d) {
        float4 v = kp[d];
        acc += v.x * v.x + v.y * v.y + v.z * v.z + v.w * v.w;
    }
    sal[i] = fminf(acc * (1.0f / (float)DH), 2.0f);
}

// ---------------------------------------------------------------------------
// Kernel 3: fused gated attention for one (b, h, 16-query) tile.
// ---------------------------------------------------------------------------
__global__ void __launch_bounds__(128)
k_attn(const float* __restrict__ qws, const float* __restrict__ kws,
       const float* __restrict__ vws, const float* __restrict__ sal,
       const float* __restrict__ ent, const int* __restrict__ layer_idx,
       float* __restrict__ ows, float* __restrict__ sstate_out)
{
    __shared__ float sP[16][SEQ];     // 64 KB score / prob buffer
    __shared__ float sSal[SEQ];       // 4 KB: 1.4 * sal[k]

    const int tid  = threadIdx.x;
    const int wave = tid >> 5;
    const int lane = tid & 31;
    const int half = lane >> 4;
    const int lm   = lane & 15;

    const int q0 = blockIdx.x * 16;
    const int h  = blockIdx.y;
    const int b  = blockIdx.z;
    const int bh = b * HEADS + h;

    const float* __restrict__ Qp = qws + (size_t)bh * SEQ * DH;
    const float* __restrict__ Kp = kws + (size_t)bh * SEQ * DH;
    const float* __restrict__ Vp = vws + (size_t)bh * SEQ * DH;

    // stage scaled saliency into LDS (consumed after the phase-1 barrier)
    {
        const float* __restrict__ salp = sal + (size_t)bh * SEQ;
        #pragma unroll
        for (int i = tid; i < SEQ; i += 128)
            sSal[i] = 1.4f * salp[i];
    }

    // Hoist the full 16x64 Q strip as A fragments (16 steps of K=4).
    v2f qa[16];
    #pragma unroll
    for (int i = 0; i < 16; ++i)
        qa[i] = *(const v2f*)(Qp + (size_t)(q0 + lm) * DH + 4 * i + 2 * half);

    // ---- Phase 1: scores into LDS; each wave covers 256 keys, 16 tiles,
    //      double-buffering the 16 K-fragments of the next tile. ----
    {
        const int ktBase = wave * 16;
        v2f fA[16], fB[16];
        #pragma unroll
        for (int i = 0; i < 16; ++i)
            fA[i] = *(const v2f*)(Kp + (size_t)(ktBase * 16 + lm) * DH + 4 * i + 2 * half);

        #pragma unroll
        for (int kt = 0; kt < 16; kt += 2) {
            // prefetch tile kt+1
            #pragma unroll
            for (int i = 0; i < 16; ++i)
                fB[i] = *(const v2f*)(Kp + (size_t)((ktBase + kt + 1) * 16 + lm) * DH + 4 * i + 2 * half);
            v8f c = {};
            #pragma unroll
            for (int i = 0; i < 16; ++i) c = WMMA_F32(qa[i], fA[i], c);
            #pragma unroll
            for (int j = 0; j < 8; ++j)
                sP[j + 8 * half][(ktBase + kt) * 16 + lm] = c[j] * 0.125f;

            // prefetch tile kt+2
            if (kt + 2 < 16) {
                #pragma unroll
                for (int i = 0; i < 16; ++i)
                    fA[i] = *(const v2f*)(Kp + (size_t)((ktBase + kt + 2) * 16 + lm) * DH + 4 * i + 2 * half);
            }
            v8f c2 = {};
            #pragma unroll
            for (int i = 0; i < 16; ++i) c2 = WMMA_F32(qa[i], fB[i], c2);
            #pragma unroll
            for (int j = 0; j < 8; ++j)
                sP[j + 8 * half][(ktBase + kt + 1) * 16 + lm] = c2[j] * 0.125f;
        }
    }
    __syncthreads();

    // ---- Phase 2: gating + masked softmax + entropy, 8 lanes per query ----
    {
        int li = layer_idx[0]; li = li < 0 ? 0 : (li > 11 ? 11 : li);
        const float tau = c_tau[li];
        const int q = tid >> 3, sub = tid & 7;
        const float NEG_INF = -__builtin_inff();

        const float e0 = ent[(size_t)bh * SEQ + q0 + q];
        const float Et = 1.f / (1.f + __expf(-5.0f * (e0 - 0.3f)));   // sigmoid(K_SCALE*(e-C))
        const float cterm = 5.0f * Et - 0.3f * e0;                    // E_t*5 - BETA*e

        // Pass A: gated score (stored in place), running max, raw argmax.
        float mx = NEG_INF, bestv = NEG_INF;
        int   besti = SEQ;
        for (int k = sub; k < SEQ; k += 8) {
            const float s  = sP[q][k];
            float gl = cterm + sSal[k];
            gl = fminf(fmaxf(gl, -20.f), 20.f);
            const float gate = 1.f / (1.f + __expf(-gl));
            const float sg = (gate >= tau) ? (s + __logf(gate + 1e-8f)) : NEG_INF;
            mx = fmaxf(mx, sg);
            if (s > bestv || (s == bestv && k < besti)) { bestv = s; besti = k; }
            sP[q][k] = sg;
        }
        #pragma unroll
        for (int off = 4; off; off >>= 1) {
            mx = fmaxf(mx, __shfl_xor(mx, off, 8));
            const float ov = __shfl_xor(bestv, off, 8);
            const int   oi = __shfl_xor(besti, off, 8);
            if (ov > bestv || (ov == bestv && oi < besti)) { bestv = ov; besti = oi; }
        }
        const bool anyf = (mx != NEG_INF);

        // Pass B: exp / sums (single exp per key; gate math already done).
        float l = 0.f, wsum = 0.f;
        for (int k = sub; k < SEQ; k += 8) {
            const float sg = sP[q][k];
            float p = 0.f;
            if (sg != NEG_INF) {
                const float shifted = sg - mx;
                p = __expf(shifted);
                l += p;
                wsum += p * shifted;
            }
            sP[q][k] = p;
        }
        #pragma unroll
        for (int off = 4; off; off >>= 1) {
            l    += __shfl_xor(l, off, 8);
            wsum += __shfl_xor(wsum, off, 8);
        }

        float scale = 1.f, sst = 0.f;
        if (anyf) {
            l = fmaxf(l, 1e-20f);
            sst = __logf(l) - wsum / l;
            scale = 1.f / l;
        }
        if (sub == 0) sstate_out[(size_t)bh * SEQ + q0 + q] = sst;

        for (int k = sub; k < SEQ; k += 8)
            sP[q][k] = anyf ? sP[q][k] * scale : ((k == besti) ? 1.f : 0.f);
    }
    __syncthreads();

    // ---- Phase 3: O = P * V ; each wave owns 16 columns of Dh; pipelined ----
    {
        const int d0 = wave * 16;
        v8f o = {};
        v2f af, bf;
        {
            const int kk = 2 * half;
            af = *(const v2f*)(&sP[lm][kk]);
            bf.x = Vp[(size_t)kk * DH + d0 + lm];
            bf.y = Vp[(size_t)(kk + 1) * DH + d0 + lm];
        }
        #pragma unroll 4
        for (int k0 = 0; k0 < SEQ - 4; k0 += 4) {
            const int kk = k0 + 4 + 2 * half;
            v2f naf = *(const v2f*)(&sP[lm][kk]);
            v2f nbf;
            nbf.x = Vp[(size_t)kk * DH + d0 + lm];
            nbf.y = Vp[(size_t)(kk + 1) * DH + d0 + lm];
            o = WMMA_F32(af, bf, o);
            af = naf; bf = nbf;
        }
        o = WMMA_F32(af, bf, o);

        #pragma unroll
        for (int j = 0; j < 8; ++j)
            ows[(size_t)(b * SEQ + q0 + j + 8 * half) * D_MODEL + h * DH + d0 + lm] = o[j];
    }
}

// ---------------------------------------------------------------------------
// Kernel 4: out = O * Wo^T  (2048x1024 * 1024x1024), row-major store
// ---------------------------------------------------------------------------
__global__ void __launch_bounds__(32)
k_out_gemm(const float* __restrict__ ows, const float* __restrict__ Wo,
           float* __restrict__ out)
{
    const int lane = threadIdx.x & 31;
    const int half = lane >> 4;
    const int lm   = lane & 15;
    const int m0   = blockIdx.x * 32;
    const int n0   = blockIdx.y * 64;

    const float* a0p = ows + (size_t)(m0 +  0 + lm) * D_MODEL + 2 * half;
    const float* a1p = ows + (size_t)(m0 + 16 + lm) * D_MODEL + 2 * half;
    const float* b0p = Wo + (size_t)(n0 +  0 + lm) * D_MODEL + 2 * half;
    const float* b1p = Wo + (size_t)(n0 + 16 + lm) * D_MODEL + 2 * half;
    const float* b2p = Wo + (size_t)(n0 + 32 + lm) * D_MODEL + 2 * half;
    const float* b3p = Wo + (size_t)(n0 + 48 + lm) * D_MODEL + 2 * half;

    v8f acc[8] = {};
    gemm32x64_pipe(a0p, a1p, b0p, b1p, b2p, b3p, acc);

    #pragma unroll
    for (int r = 0; r < 2; ++r) {
        #pragma unroll
        for (int j = 0; j < 8; ++j) {
            const size_t row = (size_t)(m0 + r * 16 + j + 8 * half) * D_MODEL;
            out[row + n0 +  0 + lm] = acc[r * 4 + 0][j];
            out[row + n0 + 16 + lm] = acc[r * 4 + 1][j];
            out[row + n0 + 32 + lm] = acc[r * 4 + 2][j];
            out[row + n0 + 48 + lm] = acc[r * 4 + 3][j];
        }
    }
}

// ---------------------------------------------------------------------------
extern "C" void kernel_launch(void* const* d_in, const int* in_sizes, int n_in,
                              void* d_out, int out_size, void* d_ws, size_t ws_size,
                              hipStream_t stream)
{
    (void)in_sizes; (void)n_in; (void)out_size; (void)ws_size;

    const float* x   = (const float*)d_in[0];
    const float* ent = (const float*)d_in[1];
    const float* Wq  = (const float*)d_in[2];
    const float* Wk  = (const float*)d_in[3];
    const float* Wv  = (const float*)d_in[4];
    const float* Wo  = (const float*)d_in[5];
    const int* layer = (const int*)d_in[6];

    float* ws   = (float*)d_ws;
    float* qws  = ws;
    float* kws  = ws + (size_t)NQ;
    float* vws  = ws + (size_t)2 * NQ;
    float* ows  = ws + (size_t)3 * NQ;     // (B,S,D)
    float* salp = ws + (size_t)4 * NQ;     // BHS floats

    float* out    = (float*)d_out;
    float* sstate = out + (size_t)BATCH * SEQ * D_MODEL;

    dim3 g1(M_TOT / 32, D_MODEL / 64, 3);
    k_qkv_gemm<<<g1, 32, 0, stream>>>(x, Wq, Wk, Wv, ws);

    k_sal<<<(BHS + 255) / 256, 256, 0, stream>>>(kws, salp);

    dim3 g2(SEQ / 16, HEADS, BATCH);
    k_attn<<<g2, 128, 0, stream>>>(qws, kws, vws, salp, ent, layer, ows, sstate);

    dim3 g3(M_TOT / 32, D_MODEL / 64);
    k_out_gemm<<<g3, 32, 0, stream>>>(ows, Wo, out);
}